// StructureEmbeddingLayer_44444321579169
// MI455X (gfx1250) — compile-verified
//
#include <hip/hip_runtime.h>

// ---------------------------------------------------------------------------
// StructureEmbeddingLayer for MI455X (gfx1250, wave32).
// Memory-bound: 315MB output stream dominates; W_rev GEMM done with
// V_WMMA_F32_16X16X4_F32 (fp32 WMMA -> exact numerics, tiny FLOP count).
// ---------------------------------------------------------------------------

namespace {
constexpr int D_ = 256;
constexpr int B_ = 128;
constexpr int N_ = 48;
constexpr int M_ = 49;   // N + 1 virtual node
constexpr int E_ = 64;
constexpr int NBOND = B_ * E_;          // 8192
constexpr float EPS_ = 1e-5f;
}

typedef __attribute__((ext_vector_type(4))) float f4;
typedef __attribute__((ext_vector_type(2))) float v2f;
typedef __attribute__((ext_vector_type(8))) float v8f;

__device__ __forceinline__ float hsum4(f4 a) { return a[0] + a[1] + a[2] + a[3]; }

// full wave32 reduction
__device__ __forceinline__ float wsum(float v) {
#pragma unroll
  for (int off = 16; off > 0; off >>= 1) v += __shfl_xor(v, off, 32);
  return v;
}

// LayerNorm over D=256 held as 8 floats/lane (two float4s at off0/off1).
__device__ __forceinline__ void ln8(f4& x0, f4& x1,
                                    const float* __restrict__ g,
                                    const float* __restrict__ be,
                                    int off0, int off1) {
  float mu = wsum(hsum4(x0) + hsum4(x1)) * (1.0f / (float)D_);
  x0 -= mu; x1 -= mu;
  float var = wsum(hsum4(x0 * x0) + hsum4(x1 * x1)) * (1.0f / (float)D_);
  float r = rsqrtf(var + EPS_);
  x0 = x0 * r * (*(const f4*)(g + off0)) + (*(const f4*)(be + off0));
  x1 = x1 * r * (*(const f4*)(g + off1)) + (*(const f4*)(be + off1));
}

// ---------------------------------------------------------------------------
// Kernel 1: hb[p,:] = (LN(sum cate rows) + LN(x@W_bf+b)) * mask ; one wave/bond
// ---------------------------------------------------------------------------
__global__ void seb_bond_embed(const int* __restrict__ bond_feat_cate,   // [B,E,3]
                               const float* __restrict__ bond_feat_float,// [B,E,4]
                               const float* __restrict__ bond_mask,      // [B,E]
                               const float* __restrict__ table_bc,       // [33,256]
                               const float* __restrict__ ln_bc_g,
                               const float* __restrict__ ln_bc_b,
                               const float* __restrict__ W_bf,           // [4,256]
                               const float* __restrict__ b_bf,
                               const float* __restrict__ ln_bf_g,
                               const float* __restrict__ ln_bf_b,
                               float* __restrict__ hb)                   // [8192,256]
{
  const int wave = threadIdx.x >> 5;
  const int lane = threadIdx.x & 31;
  const int p = blockIdx.x * 8 + wave;
  if (p >= NBOND) return;
  const int off0 = lane * 4, off1 = 128 + lane * 4;

  // categorical: sum of 3 embedding rows
  const int starts[3] = {0, 11, 22};
  f4 c0 = {0.f, 0.f, 0.f, 0.f}, c1 = {0.f, 0.f, 0.f, 0.f};
#pragma unroll
  for (int f = 0; f < 3; ++f) {
    const int row = bond_feat_cate[p * 3 + f] + 1 + starts[f];
    const float* tr = table_bc + row * D_;
    c0 += *(const f4*)(tr + off0);
    c1 += *(const f4*)(tr + off1);
  }
  ln8(c0, c1, ln_bc_g, ln_bc_b, off0, off1);

  // float: x(4) @ W_bf(4,256) + b_bf
  f4 y0 = *(const f4*)(b_bf + off0);
  f4 y1 = *(const f4*)(b_bf + off1);
#pragma unroll
  for (int k = 0; k < 4; ++k) {
    const float xk = bond_feat_float[p * 4 + k];
    y0 += xk * (*(const f4*)(W_bf + k * D_ + off0));
    y1 += xk * (*(const f4*)(W_bf + k * D_ + off1));
  }
  ln8(y0, y1, ln_bf_g, ln_bf_b, off0, off1);

  const float m = bond_mask[p];
  f4 o0 = (c0 + y0) * m;
  f4 o1 = (c1 + y1) * m;
  *(f4*)(hb + (long)p * D_ + off0) = o0;
  *(f4*)(hb + (long)p * D_ + off1) = o1;
}

// ---------------------------------------------------------------------------
// Kernel 2: hbr = hb @ W_rev + b_rev via V_WMMA_F32_16X16X4_F32.
// One wave per 16x16 output tile; K loop 256/4 = 64 WMMA ops.
// A 16x4 layout : a[vi] = A[lane&15][k + (lane>>4)*2 + vi]
// B 4x16 layout : b[vi] = B[k + (lane>>4)*2 + vi][lane&15]
// C/D 16x16     : c[r]  = D[r + 8*(lane>>4)][lane&15]
// ---------------------------------------------------------------------------
__global__ void seb_rev_gemm(const float* __restrict__ hb,   // [8192,256]
                             const float* __restrict__ Wr,   // [256,256]
                             const float* __restrict__ br,   // [256]
                             float* __restrict__ hbr)        // [8192,256]
{
  const int wave = threadIdx.x >> 5;
  const int lane = threadIdx.x & 31;
  const int t = blockIdx.x * 8 + wave;
  constexpr int MT = NBOND / 16;          // 512 tiles in M
  if (t >= MT * (D_ / 16)) return;
  const int mt = t % MT;
  const int nt = t / MT;                  // consecutive waves share B columns
  const int lo = lane & 15, hi = lane >> 4;

  v8f c;
  const float bv = br[nt * 16 + lo];
#pragma unroll
  for (int r = 0; r < 8; ++r) c[r] = bv;

  const float* arow = hb + (long)(mt * 16 + lo) * D_ + hi * 2;
  const float* bcol = Wr + (long)(hi * 2) * D_ + nt * 16 + lo;

#pragma unroll 4
  for (int k = 0; k < D_; k += 4) {
    v2f a = *(const v2f*)(arow + k);             // 8B aligned (even element)
    v2f b;
    b[0] = bcol[(long)k * D_];
    b[1] = bcol[(long)k * D_ + D_];
    c = __builtin_amdgcn_wmma_f32_16x16x4_f32(false, a, false, b,
                                              (short)0, c, false, false);
  }

  float* drow = hbr + nt * 16 + lo;
#pragma unroll
  for (int r = 0; r < 8; ++r)
    drow[(long)(mt * 16 + r + 8 * hi) * D_] = c[r];
}

// ---------------------------------------------------------------------------
// Kernel 3: full hs fill. One wave per (b,i,j) of the 49x49 grid.
// Row/col 0 -> virtual_edge_emb, interior -> fused structure block.
// Non-temporal float4 stores: 315MB streamed exactly once.
// ---------------------------------------------------------------------------
__global__ void seb_structure_fill(const int* __restrict__ sf_cate,    // [B,N,N,4]
                                   const float* __restrict__ sf_float, // [B,N,N,8]
                                   const float* __restrict__ table_sc, // [44,256]
                                   const float* __restrict__ ln_sc_g,
                                   const float* __restrict__ ln_sc_b,
                                   const float* __restrict__ W_sf,     // [8,256]
                                   const float* __restrict__ b_sf,
                                   const float* __restrict__ ln_sf_g,
                                   const float* __restrict__ ln_sf_b,
                                   const float* __restrict__ vedge,    // [256]
                                   float* __restrict__ out)            // [B,M,M,256]
{
  const int wave = threadIdx.x >> 5;
  const int lane = threadIdx.x & 31;
  const long p = (long)blockIdx.x * 8 + wave;
  constexpr long P = (long)B_ * M_ * M_;
  if (p >= P) return;

  const int rem = (int)(p % (M_ * M_));
  const int b = (int)(p / (M_ * M_));
  const int i = rem / M_;
  const int j = rem % M_;
  const int off0 = lane * 4, off1 = 128 + lane * 4;
  float* dst = out + p * D_;   // p == ((b*M + i)*M + j) by construction

  if (i == 0 || j == 0) {
    __builtin_nontemporal_store(*(const f4*)(vedge + off0), (f4*)(dst + off0));
    __builtin_nontemporal_store(*(const f4*)(vedge + off1), (f4*)(dst + off1));
    return;
  }
  const long q = ((long)(b * N_ + (i - 1)) * N_ + (j - 1));

  const int starts[4] = {0, 11, 22, 33};
  f4 c0 = {0.f, 0.f, 0.f, 0.f}, c1 = {0.f, 0.f, 0.f, 0.f};
#pragma unroll
  for (int f = 0; f < 4; ++f) {
    const int row = sf_cate[q * 4 + f] + 1 + starts[f];
    const float* tr = table_sc + row * D_;
    c0 += *(const f4*)(tr + off0);
    c1 += *(const f4*)(tr + off1);
  }
  ln8(c0, c1, ln_sc_g, ln_sc_b, off0, off1);

  const float* xf = sf_float + q * 8;
  f4 y0 = *(const f4*)(b_sf + off0);
  f4 y1 = *(const f4*)(b_sf + off1);
#pragma unroll
  for (int k = 0; k < 8; ++k) {
    const float xk = xf[k];
    y0 += xk * (*(const f4*)(W_sf + k * D_ + off0));
    y1 += xk * (*(const f4*)(W_sf + k * D_ + off1));
  }
  ln8(y0, y1, ln_sf_g, ln_sf_b, off0, off1);

  __builtin_nontemporal_store(c0 + y0, (f4*)(dst + off0));
  __builtin_nontemporal_store(c1 + y1, (f4*)(dst + off1));
}

// ---------------------------------------------------------------------------
// Kernel 4: scatter-add hb into (i,j) and hbr into (j,i). One wave per bond.
// f32 hardware atomics (duplicate bond endpoints are possible).
// ---------------------------------------------------------------------------
__global__ void seb_scatter(const int* __restrict__ bond_index, // [B,2,E]
                            const float* __restrict__ hb,
                            const float* __restrict__ hbr,
                            float* __restrict__ out)
{
  const int wave = threadIdx.x >> 5;
  const int lane = threadIdx.x & 31;
  const int p = blockIdx.x * 8 + wave;
  if (p >= NBOND) return;
  const int b = p / E_, e = p % E_;
  const int i = bond_index[(b * 2 + 0) * E_ + e] + 1;
  const int j = bond_index[(b * 2 + 1) * E_ + e] + 1;
  float* dij = out + ((long)(b * M_ + i) * M_ + j) * D_;
  float* dji = out + ((long)(b * M_ + j) * M_ + i) * D_;
  const float* s1 = hb + (long)p * D_;
  const float* s2 = hbr + (long)p * D_;
#pragma unroll
  for (int u = 0; u < 8; ++u) {
    const int d = u * 32 + lane;
    unsafeAtomicAdd(dij + d, s1[d]);   // global_atomic_add_f32
    unsafeAtomicAdd(dji + d, s2[d]);
  }
}

// ---------------------------------------------------------------------------
extern "C" void kernel_launch(void* const* d_in, const int* in_sizes, int n_in,
                              void* d_out, int out_size, void* d_ws, size_t ws_size,
                              hipStream_t stream) {
  const int*   bond_index       = (const int*)  d_in[0];
  const int*   bond_feat_cate   = (const int*)  d_in[1];
  const float* bond_feat_float  = (const float*)d_in[2];
  const float* bond_mask        = (const float*)d_in[3];
  const int*   sf_cate          = (const int*)  d_in[4];
  const float* sf_float         = (const float*)d_in[5];
  const float* table_bc         = (const float*)d_in[6];
  const float* ln_bc_g          = (const float*)d_in[7];
  const float* ln_bc_b          = (const float*)d_in[8];
  const float* W_bf             = (const float*)d_in[9];
  const float* b_bf             = (const float*)d_in[10];
  const float* ln_bf_g          = (const float*)d_in[11];
  const float* ln_bf_b          = (const float*)d_in[12];
  const float* table_sc         = (const float*)d_in[13];
  const float* ln_sc_g          = (const float*)d_in[14];
  const float* ln_sc_b          = (const float*)d_in[15];
  const float* W_sf             = (const float*)d_in[16];
  const float* b_sf             = (const float*)d_in[17];
  const float* ln_sf_g          = (const float*)d_in[18];
  const float* ln_sf_b          = (const float*)d_in[19];
  const float* vedge            = (const float*)d_in[20];
  const float* W_rev            = (const float*)d_in[21];
  const float* b_rev            = (const float*)d_in[22];
  float* out = (float*)d_out;

  float* hb  = (float*)d_ws;                       // 8192*256 f32 = 8 MB
  float* hbr = hb + (size_t)NBOND * D_;            // 8 MB more

  const dim3 blk(256);

  // 1) bond embeddings (8192 waves)
  seb_bond_embed<<<NBOND / 8, blk, 0, stream>>>(
      bond_feat_cate, bond_feat_float, bond_mask, table_bc, ln_bc_g, ln_bc_b,
      W_bf, b_bf, ln_bf_g, ln_bf_b, hb);

  // 2) reversed projection GEMM via f32 WMMA (8192 tiles / 8 waves per block)
  seb_rev_gemm<<<(NBOND / 16) * (D_ / 16) / 8, blk, 0, stream>>>(
      hb, W_rev, b_rev, hbr);

  // 3) fill entire hs (virtual edges + structure block), 307328 waves
  {
    const long P = (long)B_ * M_ * M_;
    seb_structure_fill<<<(unsigned)((P + 7) / 8), blk, 0, stream>>>(
        sf_cate, sf_float, table_sc, ln_sc_g, ln_sc_b,
        W_sf, b_sf, ln_sf_g, ln_sf_b, vedge, out);
  }

  // 4) scatter-add bond contributions
  seb_scatter<<<NBOND / 8, blk, 0, stream>>>(bond_index, hb, hbr, out);
}